// GINBlock_90580860273090
// MI455X (gfx1250) — compile-verified
//
#include <hip/hip_runtime.h>
#include <hip/hip_bf16.h>

typedef float v2f __attribute__((ext_vector_type(2)));
typedef float v8f __attribute__((ext_vector_type(8)));

#define GIN_N    40000
#define GIN_E    640000
#define GIN_DIN  128
#define GIN_DOUT 256

// ---------------------------------------------------------------------------
// Phase 1: CSR-style aggregation (avoids 82M fp32 atomics -> ~640K int atomics)
// ---------------------------------------------------------------------------

__global__ void zero_kernel(int* __restrict__ p, int n) {
    int i = blockIdx.x * 256 + threadIdx.x;
    if (i < n) p[i] = 0;
}

__global__ void count_kernel(const int* __restrict__ dst, int* __restrict__ counts, int e) {
    int i = blockIdx.x * 256 + threadIdx.x;
    if (i < e) atomicAdd(&counts[dst[i]], 1);
}

// single-block exclusive scan over n=40000 counters (cost is negligible vs. the
// fp32 atomics it eliminates)
__global__ void scan_kernel(const int* __restrict__ counts, int* __restrict__ offsets, int n) {
    __shared__ int sm[256];
    __shared__ int running;
    if (threadIdx.x == 0) running = 0;
    __syncthreads();
    for (int base = 0; base < n; base += 256) {
        int i = base + (int)threadIdx.x;
        int val = (i < n) ? counts[i] : 0;
        sm[threadIdx.x] = val;
        __syncthreads();
        for (int off = 1; off < 256; off <<= 1) {
            int t = (threadIdx.x >= (unsigned)off) ? sm[threadIdx.x - off] : 0;
            __syncthreads();
            sm[threadIdx.x] += t;
            __syncthreads();
        }
        if (i < n) offsets[i] = running + sm[threadIdx.x] - val;
        __syncthreads();
        if (threadIdx.x == 0) running += sm[255];
        __syncthreads();
    }
    if (threadIdx.x == 0) offsets[n] = running;
}

__global__ void fill_kernel(const int* __restrict__ src, const int* __restrict__ dst,
                            const int* __restrict__ offsets, int* __restrict__ cursor,
                            int* __restrict__ bucket, int e) {
    int i = blockIdx.x * 256 + threadIdx.x;
    if (i < e) {
        int d = dst[i];
        int pos = offsets[d] + atomicAdd(&cursor[d], 1);
        bucket[pos] = src[i];
    }
}

// one wave per destination node; lane covers 4 consecutive floats (float4 => 512B
// coalesced row reads). Also folds in the (1+eps)*x self term.
__global__ __launch_bounds__(256)
void gather_kernel(const float* __restrict__ x, const int* __restrict__ offsets,
                   const int* __restrict__ bucket, const float* __restrict__ epsp,
                   float* __restrict__ h) {
    int node = blockIdx.x * 8 + (threadIdx.x >> 5);
    int lane = threadIdx.x & 31;
    float se = 1.0f + epsp[0];
    const float4* xv = (const float4*)x;
    float4 self = xv[(size_t)node * 32 + lane];
    float ax = self.x * se, ay = self.y * se, az = self.z * se, aw = self.w * se;
    int e0 = offsets[node], e1 = offsets[node + 1];
    for (int i = e0; i < e1; ++i) {
        int s = bucket[i];
        float4 r = xv[(size_t)s * 32 + lane];
        ax += r.x; ay += r.y; az += r.z; aw += r.w;
    }
    float4 o; o.x = ax; o.y = ay; o.z = az; o.w = aw;
    ((float4*)h)[(size_t)node * 32 + lane] = o;
}

// ---------------------------------------------------------------------------
// Phase 2: fp32 WMMA GEMM (V_WMMA_F32_16X16X4_F32) with fused BN/ReLU/residual
// Block = 128 threads (4 waves), block tile 64x64, wave tile 32x32.
// LDS: As[64][K+4] (4-bank rotation per row -> conflict-free strided b64 reads)
//      Bs[K][72]   (2*72 mod 64 == 16 -> half-waves hit disjoint banks)
// ---------------------------------------------------------------------------

template <int K, bool USE_BN, bool USE_RES>
__global__ __launch_bounds__(128)
void gemm_kernel(const float* __restrict__ A, const float* __restrict__ B,
                 const float* __restrict__ bias, const float* __restrict__ gam,
                 const float* __restrict__ bet, const float* __restrict__ mu,
                 const float* __restrict__ var, const float* __restrict__ res,
                 float* __restrict__ out, int ncols) {
    constexpr int LDA = K + 4;
    constexpr int LDB = 72;
    extern __shared__ float lds[];
    float* As = lds;
    float* Bs = lds + 64 * LDA;

    const int rowBase = blockIdx.x * 64;
    const int colBase = blockIdx.y * 64;
    const int tid = threadIdx.x;

    // cooperative stage of A tile [64 x K]
    for (int idx = tid; idx < 64 * K / 4; idx += 128) {
        int flat = idx * 4;
        int r = flat / K, c = flat % K;
        *(float4*)(As + r * LDA + c) =
            *(const float4*)(A + (size_t)(rowBase + r) * K + c);
    }
    // cooperative stage of B panel [K x 64]
    for (int idx = tid; idx < K * 64 / 4; idx += 128) {
        int flat = idx * 4;
        int r = flat / 64, c = flat % 64;
        *(float4*)(Bs + r * LDB + c) =
            *(const float4*)(B + (size_t)r * ncols + colBase + c);
    }
    __syncthreads();

    const int wave = tid >> 5;
    const int lane = tid & 31;
    const int mrow = lane & 15;   // M (for A) / N (for B,C) within tile
    const int kh   = lane >> 4;   // half-wave selects K pair / M+8 rows
    const int rowOff = (wave >> 1) * 32;
    const int colOff = (wave & 1) * 32;

    v8f acc[2][2] = {};
    const float* Ap0 = As + (rowOff + mrow) * LDA + 2 * kh;
    const float* Ap1 = Ap0 + 16 * LDA;
    const float* Bp  = Bs + 2 * kh * LDB + colOff + mrow;

#pragma unroll 4
    for (int k = 0; k < K; k += 4) {
        v2f a0 = *(const v2f*)(Ap0 + k);   // A rows: K = 2kh, 2kh+1 (contiguous b64)
        v2f a1 = *(const v2f*)(Ap1 + k);
        const float* bk = Bp + k * LDB;
        v2f b0, b1;
        b0.x = bk[0];        // B row k+2kh
        b0.y = bk[LDB];      // B row k+2kh+1
        b1.x = bk[16];
        b1.y = bk[LDB + 16];
        acc[0][0] = __builtin_amdgcn_wmma_f32_16x16x4_f32(false, a0, false, b0, (short)0, acc[0][0], false, false);
        acc[0][1] = __builtin_amdgcn_wmma_f32_16x16x4_f32(false, a0, false, b1, (short)0, acc[0][1], false, false);
        acc[1][0] = __builtin_amdgcn_wmma_f32_16x16x4_f32(false, a1, false, b0, (short)0, acc[1][0], false, false);
        acc[1][1] = __builtin_amdgcn_wmma_f32_16x16x4_f32(false, a1, false, b1, (short)0, acc[1][1], false, false);
    }

    // epilogue: C/D layout -> VGPR r holds row (r + 8*kh), col = mrow within 16-tile
#pragma unroll
    for (int ct = 0; ct < 2; ++ct) {
        int c = colBase + colOff + ct * 16 + mrow;
        float scale = 1.0f, shift = 0.0f;
        if (USE_BN) {
            float inv = rsqrtf(var[c] + 1e-5f);
            scale = gam[c] * inv;
            shift = (bias[c] - mu[c]) * scale + bet[c];  // bias folded into BN shift
        }
#pragma unroll
        for (int rt = 0; rt < 2; ++rt) {
            int row0 = rowBase + rowOff + rt * 16 + kh * 8;
#pragma unroll
            for (int r = 0; r < 8; ++r) {
                float val = acc[rt][ct][r];
                if (USE_BN) val = val * scale + shift;
                int row = row0 + r;
                if (USE_RES) val += res[(size_t)row * ncols + c];
                val = fmaxf(val, 0.0f);
                out[(size_t)row * ncols + c] = val;
            }
        }
    }
}

// ---------------------------------------------------------------------------

extern "C" void kernel_launch(void* const* d_in, const int* in_sizes, int n_in,
                              void* d_out, int out_size, void* d_ws, size_t ws_size,
                              hipStream_t stream) {
    constexpr int N = GIN_N, E = GIN_E, DIN = GIN_DIN, DOUT = GIN_DOUT;

    const float* x    = (const float*)d_in[0];
    const int*   eidx = (const int*)d_in[1];
    const float* eps  = (const float*)d_in[2];
    const float* W1   = (const float*)d_in[3];
    const float* b1   = (const float*)d_in[4];
    const float* g1   = (const float*)d_in[5];
    const float* bt1  = (const float*)d_in[6];
    const float* m1   = (const float*)d_in[7];
    const float* v1   = (const float*)d_in[8];
    const float* W2   = (const float*)d_in[9];
    const float* b2   = (const float*)d_in[10];
    const float* g2   = (const float*)d_in[11];
    const float* bt2  = (const float*)d_in[12];
    const float* m2   = (const float*)d_in[13];
    const float* v2   = (const float*)d_in[14];
    const float* Wres = (const float*)d_in[15];

    const int* src = eidx;       // edge_index[0]
    const int* dst = eidx + E;   // edge_index[1]

    // workspace layout
    float* h  = (float*)d_ws;                    // N*DIN
    float* h1 = h + (size_t)N * DIN;             // N*DOUT
    float* h2 = h1 + (size_t)N * DOUT;           // N*DOUT
    int* counts  = (int*)(h2 + (size_t)N * DOUT); // N
    int* cursor  = counts + N;                    // N
    int* offsets = cursor + N;                    // N+1
    int* bucket  = offsets + N + 1;               // E

    // ---- aggregation (CSR build + gather) ----
    zero_kernel<<<(2 * N + 255) / 256, 256, 0, stream>>>(counts, 2 * N);
    count_kernel<<<E / 256, 256, 0, stream>>>(dst, counts, E);
    scan_kernel<<<1, 256, 0, stream>>>(counts, offsets, N);
    fill_kernel<<<E / 256, 256, 0, stream>>>(src, dst, offsets, cursor, bucket, E);
    gather_kernel<<<N / 8, 256, 0, stream>>>(x, offsets, bucket, eps, h);

    // ---- GEMM chain (fp32 WMMA, fused epilogues) ----
    dim3 grid(N / 64, DOUT / 64);
    size_t lds128 = (size_t)(64 * (128 + 4) + 128 * 72) * 4;  // 70,656 B
    size_t lds256 = (size_t)(64 * (256 + 4) + 256 * 72) * 4;  // 140,288 B

    // h1 = relu(bn1(h @ W1 + b1))
    gemm_kernel<128, true, false><<<grid, 128, lds128, stream>>>(
        h, W1, b1, g1, bt1, m1, v1, nullptr, h1, DOUT);
    // h2 = relu(bn2(h1 @ W2 + b2))
    gemm_kernel<256, true, false><<<grid, 128, lds256, stream>>>(
        h1, W2, b2, g2, bt2, m2, v2, nullptr, h2, DOUT);
    // out = relu(h2 + x @ Wres)
    gemm_kernel<128, false, true><<<grid, 128, lds128, stream>>>(
        x, Wres, nullptr, nullptr, nullptr, nullptr, nullptr, h2, (float*)d_out, DOUT);
}